// ModelNew_89103391523416
// MI455X (gfx1250) — compile-verified
//
#include <hip/hip_runtime.h>

// Row-wise cumulative product (fp32), 4096 rows x 32768 cols.
// Bandwidth-bound scan: ~1.07 GB traffic -> ~46 us floor at 23.3 TB/s.
// One wave32 per row; segmented scan: per-lane serial product (16 elems),
// wave scan via ds_bpermute (5 steps), broadcast via v_readlane, b128
// non-temporal streaming, global_prefetch_b8 lookahead.

#define COLS            32768
#define VPL             16                 // elements per lane per tile
#define TILE            (32 * VPL)         // 512 elements per wave per tile
#define ITERS           (COLS / TILE)      // 64 tiles per row
#define WAVES_PER_BLOCK 8
#define PREFETCH_TILES  4                  // lookahead distance (tiles)

typedef __attribute__((ext_vector_type(4))) float v4f;

// Wave32 lane gather: dst[lane] = v[src_lane]. Lowers to ds_bpermute_b32
// (LDS permute network, no LDS memory, no barrier). Negative src_lane wraps
// in hardware (index bits [6:2]); callers mask those lanes out.
__device__ __forceinline__ float bperm_f(int src_lane, float v) {
    return __int_as_float(
        __builtin_amdgcn_ds_bpermute(src_lane << 2, __float_as_int(v)));
}

// Wave-uniform broadcast of lane 31 via v_readlane_b32 (cheaper than a
// ds_bpermute round-trip; result lands in an SGPR).
__device__ __forceinline__ float readlane31_f(float v) {
    return __int_as_float(__builtin_amdgcn_readlane(__float_as_int(v), 31));
}

__global__ __launch_bounds__(WAVES_PER_BLOCK * 32)
void cumprod_rows_kernel(const float* __restrict__ x,
                         float* __restrict__ y,
                         int rows) {
    const int lane = threadIdx.x & 31;
    const int wave = threadIdx.x >> 5;
    const int row  = blockIdx.x * WAVES_PER_BLOCK + wave;
    if (row >= rows) return;

    const float* __restrict__ xp = x + (size_t)row * COLS + (size_t)lane * VPL;
    float*       __restrict__ yp = y + (size_t)row * COLS + (size_t)lane * VPL;

    float carry = 1.0f;  // running product of everything left of this tile

    #pragma unroll 2
    for (int t = 0; t < ITERS; ++t) {
        // Prefetch a future tile ahead of the serial carry chain.
        // Emits global_prefetch_b8 on gfx1250.
        if (t + PREFETCH_TILES < ITERS) {
            __builtin_prefetch(xp + (size_t)PREFETCH_TILES * TILE, 0, 1);
        }

        // Streamed-once data: non-temporal b128 loads (don't pollute 192MB L2).
        v4f a = __builtin_nontemporal_load((const v4f*)xp);
        v4f b = __builtin_nontemporal_load((const v4f*)(xp + 4));
        v4f c = __builtin_nontemporal_load((const v4f*)(xp + 8));
        v4f d = __builtin_nontemporal_load((const v4f*)(xp + 12));

        // 1) Per-lane inclusive product of its 16 contiguous elements.
        a.y *= a.x;  a.z *= a.y;  a.w *= a.z;
        b.x *= a.w;  b.y *= b.x;  b.z *= b.y;  b.w *= b.z;
        c.x *= b.w;  c.y *= c.x;  c.z *= c.y;  c.w *= c.z;
        d.x *= c.w;  d.y *= d.x;  d.z *= d.y;  d.w *= d.z;
        const float lane_total = d.w;

        // 2) Wave-wide inclusive scan (product) of lane totals:
        //    5 ds_bpermute + select-multiply steps (wave32).
        float s = lane_total;
        #pragma unroll
        for (int dd = 1; dd < 32; dd <<= 1) {
            float up = bperm_f(lane - dd, s);
            s = (lane >= dd) ? s * up : s;
        }

        // 3) Exclusive prefix for this lane (lane 0 -> 1.0), times row carry.
        float excl = bperm_f(lane - 1, s);
        excl = (lane == 0) ? 1.0f : excl;
        const float pre = carry * excl;

        // Whole-wave total (uniform) advances the carry; SGPR broadcast.
        const float wave_total = readlane31_f(s);

        // 4) Scale local inclusive products into final cumprod values.
        a.x *= pre;  a.y *= pre;  a.z *= pre;  a.w *= pre;
        b.x *= pre;  b.y *= pre;  b.z *= pre;  b.w *= pre;
        c.x *= pre;  c.y *= pre;  c.z *= pre;  c.w *= pre;
        d.x *= pre;  d.y *= pre;  d.z *= pre;  d.w *= pre;

        __builtin_nontemporal_store(a, (v4f*)yp);
        __builtin_nontemporal_store(b, (v4f*)(yp + 4));
        __builtin_nontemporal_store(c, (v4f*)(yp + 8));
        __builtin_nontemporal_store(d, (v4f*)(yp + 12));

        carry *= wave_total;
        xp += TILE;
        yp += TILE;
    }
}

extern "C" void kernel_launch(void* const* d_in, const int* in_sizes, int n_in,
                              void* d_out, int out_size, void* d_ws, size_t ws_size,
                              hipStream_t stream) {
    (void)n_in; (void)out_size; (void)d_ws; (void)ws_size;

    const float* x = (const float*)d_in[0];
    float*       y = (float*)d_out;

    const int total = in_sizes[0];          // 4096 * 32768
    const int rows  = total / COLS;         // 4096

    dim3 block(WAVES_PER_BLOCK * 32);       // 256 threads = 8 wave32 = 8 rows
    dim3 grid((rows + WAVES_PER_BLOCK - 1) / WAVES_PER_BLOCK);

    cumprod_rows_kernel<<<grid, block, 0, stream>>>(x, y, rows);
}